// DeformConv_53171695125006
// MI455X (gfx1250) — compile-verified
//
#include <hip/hip_runtime.h>
#include <math.h>

// Problem constants (match reference)
#define B_    2
#define C_IN  128
#define C_OUT 256
#define H_    128
#define W_    128
#define HW    (H_ * W_)
#define NTOT  (B_ * HW)      // 32768 output pixels
#define KTAPS 9
#define NT    64             // pixels per workgroup tile
#define LDP   (NT + 1)       // padded LDS row stride (bank stagger)
#define EPS_  1e-5f
#define MOFF  32             // padded M for the offset conv (27 -> 32)

typedef __attribute__((ext_vector_type(8))) float v8f;
typedef __attribute__((ext_vector_type(2))) float v2f;

// D = A(16x4 f32) * B(4x16 f32) + C(16x16 f32), full f32 precision.
__device__ __forceinline__ v8f wmma_f32(v2f a, v2f b, v8f c) {
  return __builtin_amdgcn_wmma_f32_16x16x4_f32(
      /*neg_a=*/false, a, /*neg_b=*/false, b,
      /*c_mod=*/(short)0, c, /*reuse_a=*/false, /*reuse_b=*/false);
}

// ---------------------------------------------------------------------------
// Weight pre-pack: Wp[tap][c/2][m][c&1]  ->  A fragment = 1 coalesced b64 load
// (lane ml holds row m; lanes 0-15 read c-pair kk, lanes 16-31 pair kk+2).
// ---------------------------------------------------------------------------
__global__ __launch_bounds__(256) void pack_wdcn_kernel(
    const float* __restrict__ w, float* __restrict__ wp) {
  const int i = blockIdx.x * 256 + threadIdx.x;  // over C_OUT*C_IN*KTAPS
  if (i < C_OUT * C_IN * KTAPS) {
    const int tap = i % KTAPS;
    const int c   = (i / KTAPS) % C_IN;
    const int m   = i / (KTAPS * C_IN);
    wp[(((size_t)tap * (C_IN / 2) + (c >> 1)) * C_OUT + m) * 2 + (c & 1)] =
        w[i];
  }
}

__global__ __launch_bounds__(256) void pack_woff_kernel(
    const float* __restrict__ w, float* __restrict__ wp) {
  const int i = blockIdx.x * 256 + threadIdx.x;  // over MOFF*C_IN*KTAPS
  if (i < MOFF * C_IN * KTAPS) {
    const int tap = i % KTAPS;
    const int c   = (i / KTAPS) % C_IN;
    const int m   = i / (KTAPS * C_IN);
    const float v = (m < 27) ? w[((size_t)m * C_IN + c) * KTAPS + tap] : 0.f;
    wp[(((size_t)tap * (C_IN / 2) + (c >> 1)) * MOFF + m) * 2 + (c & 1)] = v;
  }
}

// Fold bias + BN(inference) into per-channel scale/shift:
//   out = relu(acc * sc[m] + sh[m])
__global__ __launch_bounds__(256) void bn_prep_kernel(
    const float* __restrict__ b_dcn, const float* __restrict__ gamma,
    const float* __restrict__ beta, const float* __restrict__ mean,
    const float* __restrict__ var, float* __restrict__ sc,
    float* __restrict__ sh) {
  const int m = blockIdx.x * 256 + threadIdx.x;
  if (m < C_OUT) {
    const float inv = gamma[m] * rsqrtf(var[m] + EPS_);
    sc[m] = inv;
    sh[m] = (b_dcn[m] - mean[m]) * inv + beta[m];
  }
}

// ---------------------------------------------------------------------------
// Kernel A: 3x3 SAME conv producing 27 channels (offsets + mask logits),
// fused into per-(tap,pixel) sampling coordinates py/px and sigmoid mask.
// M = 32 (27 zero-padded), K = 1152, N = 64 per workgroup.
// ---------------------------------------------------------------------------
__global__ __launch_bounds__(256) void offset_conv_kernel(
    const float* __restrict__ x, const float* __restrict__ wpo,
    const float* __restrict__ b_off, float* __restrict__ py,
    float* __restrict__ px, float* __restrict__ mk) {
  __shared__ float sval[C_IN * LDP];   // im2col tile [128ch][64px]
  __shared__ float swi[MOFF * NT];     // conv result dump [32ch][64px]

  const int tid   = threadIdx.x;
  const int lane  = tid & 31;
  const int wv    = tid >> 5;          // 8 waves
  const int mt    = wv & 1;            // m-tile: 0 or 16
  const int nt    = wv >> 1;           // n-tile: 0..3
  const int n0    = blockIdx.x * NT;
  const int b     = n0 / HW;
  const int p0    = n0 % HW;
  const int iy    = p0 / W_;           // whole tile on one image row
  const int ix0   = p0 % W_;
  const int nn    = tid & 63;
  const int c4    = tid >> 6;
  const int ml    = lane & 15;
  const int khalf = (lane >> 4) << 1;  // 0 or 2
  const int chalf = (lane >> 4);       // 0 or 1 (c-pair select)

  v8f acc = {0.f, 0.f, 0.f, 0.f, 0.f, 0.f, 0.f, 0.f};
  const int mm = mt * 16 + ml;
  const int nb = nt * 16 + ml;

  for (int tap = 0; tap < KTAPS; ++tap) {
    const int ky = tap / 3 - 1;
    const int kx = tap % 3 - 1;
    __syncthreads();  // previous GEMM done reading sval
    {  // build shifted x tile; clamped address + 0/1 multiply (no divergence)
      const int yy = iy + ky;
      const int xx = ix0 + nn + kx;
      const bool ok = (yy >= 0) & (yy < H_) & (xx >= 0) & (xx < W_);
      const float okf = ok ? 1.f : 0.f;
      const float* xb = x + (size_t)b * C_IN * HW + (ok ? (yy * W_ + xx) : 0);
      for (int c = c4; c < C_IN; c += 4)
        sval[c * LDP + nn] = xb[(size_t)c * HW] * okf;
    }
    __syncthreads();
    const v2f* ap = (const v2f*)(wpo + (size_t)tap * (C_IN / 2) * MOFF * 2);
    if (tap + 1 < KTAPS)  // warm next tap's A slice (global_prefetch_b8)
      __builtin_prefetch(
          wpo + (size_t)(tap + 1) * (C_IN / 2) * MOFF * 2 + tid * 4, 0, 0);
    // software-pipelined K loop: prefetch next fragments under current WMMA
    v2f a, bb;
    a = ap[(size_t)chalf * MOFF + mm];
    bb.x = sval[khalf * LDP + nb];
    bb.y = sval[(khalf + 1) * LDP + nb];
    for (int kk = 0; kk < C_IN - 4; kk += 4) {
      const int c2n = ((kk + 4) >> 1) + chalf;
      const int krn = kk + 4 + khalf;
      v2f an = ap[(size_t)c2n * MOFF + mm];
      v2f bn;
      bn.x = sval[krn * LDP + nb];
      bn.y = sval[(krn + 1) * LDP + nb];
      acc = wmma_f32(a, bb, acc);
      a = an;
      bb = bn;
    }
    acc = wmma_f32(a, bb, acc);
  }
  __syncthreads();
  // dump C-tile (+bias): vgpr r -> M = r (lanes0-15) / r+8 (lanes16-31)
#pragma unroll
  for (int r = 0; r < 8; ++r) {
    const int m = mt * 16 + r + ((lane >> 4) << 3);
    const float bv = (m < 27) ? b_off[m] : 0.f;
    swi[m * NT + nb] = acc[r] + bv;
  }
  __syncthreads();
  // epilogue: offset = ch 0..17 (dy=2k, dx=2k+1), mask = sigmoid(ch 18+k)
  for (int it = tid; it < KTAPS * NT; it += 256) {
    const int tap = it / NT;
    const int n   = it % NT;
    const float dy = swi[(2 * tap) * NT + n];
    const float dx = swi[(2 * tap + 1) * NT + n];
    const float mv = 1.0f / (1.0f + expf(-swi[(18 + tap) * NT + n]));
    const int gn = n0 + n;
    py[tap * NTOT + gn] = (float)(iy + tap / 3 - 1) + dy;
    px[tap * NTOT + gn] = (float)(ix0 + n + tap % 3 - 1) + dx;
    mk[tap * NTOT + gn] = mv;
  }
}

// ---------------------------------------------------------------------------
// Kernel B: modulated deformable conv as fused gather + WMMA GEMM + BN + ReLU.
// M = 256, K = 1152, N = 64 per workgroup; 8 waves x (2 M-tiles x 4 N-tiles).
// ---------------------------------------------------------------------------
__global__ __launch_bounds__(256) void dcn_kernel(
    const float* __restrict__ x, const float* __restrict__ wpd,
    const float* __restrict__ py, const float* __restrict__ px,
    const float* __restrict__ mk, const float* __restrict__ sc,
    const float* __restrict__ sh, float* __restrict__ out) {
  __shared__ float sval[C_IN * LDP];   // modulated im2col tile [128ch][64px]
  __shared__ float scw[4 * NT];        // bilinear corner weights (mask-premul)
  __shared__ int   sci[4 * NT];        // bilinear corner indices (clamped)

  const int tid   = threadIdx.x;
  const int lane  = tid & 31;
  const int wv    = tid >> 5;          // wave -> m0 = wv*32
  const int n0    = blockIdx.x * NT;
  const int b     = n0 / HW;
  const int p0    = n0 % HW;
  const int nn    = tid & 63;
  const int c4    = tid >> 6;
  const int ml    = lane & 15;
  const int khalf = (lane >> 4) << 1;
  const int chalf = (lane >> 4);
  const int m0    = wv * 32 + ml;

  const v8f vzero = {0.f, 0.f, 0.f, 0.f, 0.f, 0.f, 0.f, 0.f};
  v8f acc[2][4];
#pragma unroll
  for (int i = 0; i < 2; ++i)
#pragma unroll
    for (int j = 0; j < 4; ++j) acc[i][j] = vzero;

  for (int tap = 0; tap < KTAPS; ++tap) {
    __syncthreads();
    {  // bilinear setup: all 256 threads, one (corner, pixel) pair each
      const int cr = tid >> 6;  // corner 0..3
      const int pi = tid & 63;  // pixel in tile
      const int gn = n0 + pi;
      const float pyv = py[tap * NTOT + gn];
      const float pxv = px[tap * NTOT + gn];
      const float mv  = mk[tap * NTOT + gn];
      const float y0f = floorf(pyv), x0f = floorf(pxv);
      const float wy = pyv - y0f, wx = pxv - x0f;
      const int dyc = cr >> 1, dxc = cr & 1;
      const int yi = (int)y0f + dyc, xi = (int)x0f + dxc;
      const bool okc = (yi >= 0) & (yi < H_) & (xi >= 0) & (xi < W_);
      const int yc = min(max(yi, 0), H_ - 1);
      const int xc = min(max(xi, 0), W_ - 1);
      sci[cr * NT + pi] = yc * W_ + xc;
      const float wgt = (dyc ? wy : (1.f - wy)) * (dxc ? wx : (1.f - wx)) * mv;
      scw[cr * NT + pi] = okc ? wgt : 0.f;
    }
    __syncthreads();
    {  // gather + blend into im2col tile (x is L2-resident: 16.8MB << 192MB)
      const float* xb = x + (size_t)b * C_IN * HW;
      const int   i00 = sci[0 * NT + nn], i01 = sci[1 * NT + nn];
      const int   i10 = sci[2 * NT + nn], i11 = sci[3 * NT + nn];
      const float w00 = scw[0 * NT + nn], w01 = scw[1 * NT + nn];
      const float w10 = scw[2 * NT + nn], w11 = scw[3 * NT + nn];
      for (int c = c4; c < C_IN; c += 4) {
        const float* xc = xb + (size_t)c * HW;
        sval[c * LDP + nn] =
            w00 * xc[i00] + w01 * xc[i01] + w10 * xc[i10] + w11 * xc[i11];
      }
    }
    __syncthreads();
    const v2f* ap = (const v2f*)(wpd + (size_t)tap * (C_IN / 2) * C_OUT * 2);
    if (tap + 1 < KTAPS) {  // warm next tap's A slice + coords
      __builtin_prefetch(
          wpd + (size_t)(tap + 1) * (C_IN / 2) * C_OUT * 2 + tid * 4, 0, 0);
      __builtin_prefetch(py + (size_t)(tap + 1) * NTOT + n0 + nn, 0, 0);
      __builtin_prefetch(px + (size_t)(tap + 1) * NTOT + n0 + nn, 0, 0);
    }
    // software-pipelined K loop with register double-buffering
    v2f a0, a1, bc[4];
    a0 = ap[(size_t)chalf * C_OUT + m0];
    a1 = ap[(size_t)chalf * C_OUT + m0 + 16];
#pragma unroll
    for (int ntb = 0; ntb < 4; ++ntb) {
      bc[ntb].x = sval[khalf * LDP + ntb * 16 + ml];
      bc[ntb].y = sval[(khalf + 1) * LDP + ntb * 16 + ml];
    }
    for (int kk = 0; kk < C_IN - 4; kk += 4) {
      const int c2n = ((kk + 4) >> 1) + chalf;
      const int krn = kk + 4 + khalf;
      v2f a0n = ap[(size_t)c2n * C_OUT + m0];
      v2f a1n = ap[(size_t)c2n * C_OUT + m0 + 16];
      v2f bn[4];
#pragma unroll
      for (int ntb = 0; ntb < 4; ++ntb) {
        bn[ntb].x = sval[krn * LDP + ntb * 16 + ml];
        bn[ntb].y = sval[(krn + 1) * LDP + ntb * 16 + ml];
      }
#pragma unroll
      for (int ntb = 0; ntb < 4; ++ntb) {
        acc[0][ntb] = wmma_f32(a0, bc[ntb], acc[0][ntb]);
        acc[1][ntb] = wmma_f32(a1, bc[ntb], acc[1][ntb]);
      }
      a0 = a0n;
      a1 = a1n;
#pragma unroll
      for (int ntb = 0; ntb < 4; ++ntb) bc[ntb] = bn[ntb];
    }
#pragma unroll
    for (int ntb = 0; ntb < 4; ++ntb) {
      acc[0][ntb] = wmma_f32(a0, bc[ntb], acc[0][ntb]);
      acc[1][ntb] = wmma_f32(a1, bc[ntb], acc[1][ntb]);
    }
  }
  // fused scale/shift + ReLU epilogue; coalesced 16-wide stores
#pragma unroll
  for (int mtile = 0; mtile < 2; ++mtile) {
#pragma unroll
    for (int r = 0; r < 8; ++r) {
      const int m = wv * 32 + mtile * 16 + r + ((lane >> 4) << 3);
      const float scv = sc[m];
      const float shv = sh[m];
      float* op = out + ((size_t)b * C_OUT + m) * HW + p0 + ml;
#pragma unroll
      for (int ntb = 0; ntb < 4; ++ntb) {
        const float v = acc[mtile][ntb][r] * scv + shv;
        op[ntb * 16] = fmaxf(v, 0.f);
      }
    }
  }
}

// ---------------------------------------------------------------------------
extern "C" void kernel_launch(void* const* d_in, const int* in_sizes, int n_in,
                              void* d_out, int out_size, void* d_ws,
                              size_t ws_size, hipStream_t stream) {
  (void)in_sizes; (void)n_in; (void)out_size; (void)ws_size;
  const float* x      = (const float*)d_in[0];
  const float* w_off  = (const float*)d_in[1];
  const float* b_off  = (const float*)d_in[2];
  const float* w_dcn  = (const float*)d_in[3];
  const float* b_dcn  = (const float*)d_in[4];
  const float* gamma  = (const float*)d_in[5];
  const float* beta   = (const float*)d_in[6];
  const float* mmean  = (const float*)d_in[7];
  const float* mvar   = (const float*)d_in[8];
  float* out = (float*)d_out;

  // workspace layout (floats): py/px/mask [9][NTOT], packed weights, BN terms
  float* py  = (float*)d_ws;
  float* px  = py + (size_t)KTAPS * NTOT;
  float* mk  = px + (size_t)KTAPS * NTOT;
  float* wpd = mk + (size_t)KTAPS * NTOT;            // 256*128*9 floats
  float* wpo = wpd + (size_t)C_OUT * C_IN * KTAPS;   // 32*128*9 floats
  float* sc  = wpo + (size_t)MOFF * C_IN * KTAPS;    // 256 floats
  float* sh  = sc + C_OUT;                           // 256 floats

  const int npackd = (C_OUT * C_IN * KTAPS + 255) / 256;
  const int npacko = (MOFF * C_IN * KTAPS + 255) / 256;
  pack_wdcn_kernel<<<npackd, 256, 0, stream>>>(w_dcn, wpd);
  pack_woff_kernel<<<npacko, 256, 0, stream>>>(w_off, wpo);
  bn_prep_kernel<<<1, 256, 0, stream>>>(b_dcn, gamma, beta, mmean, mvar, sc,
                                        sh);

  const int nblocks = NTOT / NT;  // 512
  offset_conv_kernel<<<nblocks, 256, 0, stream>>>(x, wpo, b_off, py, px, mk);
  dcn_kernel<<<nblocks, 256, 0, stream>>>(x, wpd, py, px, mk, sc, sh, out);
}